// InternalCoordinateTransform_23562190586385
// MI455X (gfx1250) — compile-verified
//
#include <hip/hip_runtime.h>
#include <math.h>

#define N_ATOMS 2500
#define N_DIM   (3 * N_ATOMS)   /* 7500 floats per row  */
#define Z_ROWS  (N_ATOMS - 3)   /* 2497 z-matrix rows   */
#define THREADS 256
#define PI_F    3.14159265358979323846f

typedef unsigned int u32x4 __attribute__((ext_vector_type(4)));
typedef int          i32x8 __attribute__((ext_vector_type(8)));
typedef int          i32x4 __attribute__((ext_vector_type(4)));

struct F3 { float x, y, z; };
__device__ __forceinline__ F3    f3sub(F3 a, F3 b) { return {a.x - b.x, a.y - b.y, a.z - b.z}; }
__device__ __forceinline__ float f3dot(F3 a, F3 b) { return a.x * b.x + a.y * b.y + a.z * b.z; }
__device__ __forceinline__ F3    f3scale(F3 a, float s) { return {a.x * s, a.y * s, a.z * s}; }
__device__ __forceinline__ F3    f3cross(F3 a, F3 b) {
    return {a.y * b.z - a.z * b.y, a.z * b.x - a.x * b.z, a.x * b.y - a.y * b.x};
}

__global__ __launch_bounds__(THREADS)
void icf_kernel(const float* __restrict__ x, const int* __restrict__ zmat,
                const float* __restrict__ mb, const float* __restrict__ sb,
                const float* __restrict__ ma, const float* __restrict__ sa,
                const float* __restrict__ md, const float* __restrict__ sd,
                float* __restrict__ out)
{
    __shared__ float sx[N_DIM];            // 30 KB: one full x row
    const int    tid  = threadIdx.x;
    const size_t base = (size_t)blockIdx.x * N_DIM;
    const float* xrow = x + base;

#if defined(__HIP_DEVICE_COMPILE__) && defined(__gfx1250__) && __has_builtin(__builtin_amdgcn_tensor_load_to_lds)
    // ---- CDNA5 Tensor Data Mover: DMA the whole row (7500 f32) global -> LDS ----
    if (tid < 32) {                        // wave 0 only (branch is wave-uniform)
        // LDS flat addressing: low 32 bits of the generic pointer ARE the LDS byte offset.
        unsigned           lds_addr = (unsigned)(unsigned long long)(const void*)&sx[0];
        unsigned long long ga       = (unsigned long long)xrow;

        u32x4 g0;
        g0[0] = 1u;                                            // count=1 valid descriptor
        g0[1] = lds_addr;                                      // LDS byte address
        g0[2] = (unsigned)(ga & 0xFFFFFFFFu);                  // global_addr[31:0]
        g0[3] = (unsigned)((ga >> 32) & 0x01FFFFFFu)           // global_addr[56:32]
              | 0x80000000u;                                   // type = 2 ("image")

        i32x8 g1;
        g1[0] = (int)(2u << 16);                               // data_size=4B, workgroup_mask=0
        g1[1] = (int)(((unsigned)N_DIM & 0xFFFFu) << 16);      // tensor_dim0[15:0]
        g1[2] = (int)((((unsigned)N_DIM >> 16) & 0xFFFFu)      // tensor_dim0[31:16]
              | (1u << 16));                                   // tensor_dim1 = 1
        g1[3] = (int)((unsigned)N_DIM << 16);                  // tensor_dim1[31:16]=0, tile_dim0=7500
        g1[4] = 1;                                             // tile_dim1=1, tile_dim2=0
        g1[5] = N_DIM;                                         // tensor_dim0_stride[31:0]
        g1[6] = (int)((unsigned)N_DIM << 16);                  // stride0[47:32]=0, stride1[15:0]=7500
        g1[7] = 0;                                             // stride1[47:16]=0

        i32x4 g2; g2[0] = 1; g2[1] = 1; g2[2] = 0; g2[3] = 0;  // tensor_dim2=1, tensor_dim3=1
        i32x4 g3; g3[0] = 0; g3[1] = (int)(1u << 16);          // tensor_dim4=1
        g3[2] = 0; g3[3] = 0;

#if __clang_major__ >= 23
        i32x8 g4 = {0, 0, 0, 0, 0, 0, 0, 0};
        __builtin_amdgcn_tensor_load_to_lds(g0, g1, g2, g3, g4, 0);
#else
        __builtin_amdgcn_tensor_load_to_lds(g0, g1, g2, g3, 0);
#endif
        __builtin_amdgcn_s_wait_tensorcnt(0);                  // s_wait_tensorcnt 0
    }
    __syncthreads();
#else
    for (int i = tid; i < N_DIM; i += THREADS) sx[i] = xrow[i];
    __syncthreads();
#endif

    // ---- per-z internal coordinates out of LDS ----
    for (int z = tid; z < Z_ROWS; z += THREADS) {
        const int4 zr = ((const int4*)zmat)[z];                // {a4, a1, a2, a3}
        const int a4 = zr.x, a1 = zr.y, a2 = zr.z, a3 = zr.w;

        F3 p1 = {sx[3 * a1], sx[3 * a1 + 1], sx[3 * a1 + 2]};
        F3 p2 = {sx[3 * a2], sx[3 * a2 + 1], sx[3 * a2 + 2]};
        F3 p3 = {sx[3 * a3], sx[3 * a3 + 1], sx[3 * a3 + 2]};
        F3 p4 = {sx[3 * a4], sx[3 * a4 + 1], sx[3 * a4 + 2]};

        // bond
        F3    d41  = f3sub(p4, p1);
        float bond = sqrtf(f3dot(d41, d41));

        // angle between unit(p2-p1) and unit(p4-p1)
        F3    u21    = f3sub(p2, p1);
        float n21    = sqrtf(f3dot(u21, u21));
        float cosang = f3dot(u21, d41) / (n21 * bond);
        cosang       = fminf(1.0f, fmaxf(-1.0f, cosang));
        float ang    = acosf(cosang);

        // dihedral
        F3    b0  = f3sub(p3, p2);
        F3    b1v = f3sub(p1, p2);
        float n12 = sqrtf(f3dot(b1v, b1v));
        b1v       = f3scale(b1v, 1.0f / n12);
        F3    v   = f3sub(b0,  f3scale(b1v, f3dot(b0,  b1v)));
        F3    w   = f3sub(d41, f3scale(b1v, f3dot(d41, b1v)));
        float xx  = f3dot(v, w);
        float yy  = f3dot(f3cross(b1v, v), w);
        float dih = -atan2f(yy, xx);

        // normalize
        float bn = (bond - mb[z]) / sb[z];
        float an = (ang  - ma[z]) / sa[z];
        float d  = dih - md[z];
        d = (d < -PI_F) ? d + 2.0f * PI_F : d;
        d = (d >  PI_F) ? d - 2.0f * PI_F : d;
        float dn = d / sd[z];

        float* o = out + base + 3 * a4;     // contiguous 12B per thread -> coalesced b96
        o[0] = bn; o[1] = an; o[2] = dn;
    }

    // first 3 atoms pass through unchanged
    if (tid < 9) out[base + tid] = sx[tid];
}

extern "C" void kernel_launch(void* const* d_in, const int* in_sizes, int n_in,
                              void* d_out, int out_size, void* d_ws, size_t ws_size,
                              hipStream_t stream)
{
    const float* x    = (const float*)d_in[0];
    const int*   zmat = (const int*)  d_in[1];
    const float* mb   = (const float*)d_in[2];
    const float* sb   = (const float*)d_in[3];
    const float* ma   = (const float*)d_in[4];
    const float* sa   = (const float*)d_in[5];
    const float* md   = (const float*)d_in[6];
    const float* sd   = (const float*)d_in[7];
    float*       out  = (float*)d_out;

    const int batches = in_sizes[0] / N_DIM;   // 2048
    icf_kernel<<<batches, THREADS, 0, stream>>>(x, zmat, mb, sb, ma, sa, md, sd, out);
    (void)n_in; (void)out_size; (void)d_ws; (void)ws_size;
}